// SSIMLoss_88536455840417
// MI455X (gfx1250) — compile-verified
//
#include <hip/hip_runtime.h>

typedef __attribute__((ext_vector_type(2))) float v2f;
typedef __attribute__((ext_vector_type(8))) float v8f;

#define WAVES   4          // waves per block (1 strip per wave)
#define PSTRIDE 29         // odd LDS stride (28 cols used) -> conflict-free
#define HSTRIDE 16         // H buffer row stride (floats)
#define HQ      (32 * HSTRIDE)   // per-quantity H block (floats)
#define PROWS   26         // y0-5 .. y0+20
#define PCOLS   28         // x0-5 .. x0+22 (k=26,27 are zero-weighted)
#define IMG     512
#define NPIX    25165824.0 // 32*3*512*512

// 11-tap Gaussian, sigma=1.5, normalized (fp32)
__constant__ float GW[11] = {
    0.0010284f, 0.0075988f, 0.0360008f, 0.1093607f, 0.2130055f,
    0.2660112f,
    0.2130055f, 0.1093607f, 0.0360008f, 0.0075988f, 0.0010284f};

__device__ __forceinline__ float band(int j) {
    return (j >= 0 && j <= 10) ? GW[j] : 0.0f;
}

__device__ __forceinline__ void lds_fence() {
    // same-wave DS ops are in-order in HW; this pins the compiler and drains DScnt
    asm volatile("s_wait_dscnt 0x0" ::: "memory");
}

__global__ __launch_bounds__(WAVES * 32) void ssim_main(
    const float* __restrict__ P, const float* __restrict__ T,
    double* __restrict__ acc)
{
    __shared__ float sP[WAVES][PROWS * PSTRIDE];
    __shared__ float sT[WAVES][PROWS * PSTRIDE];
    __shared__ float sH[WAVES][5 * HQ];          // 5 quantities x 32 rows x 16 cols

    const int lane  = threadIdx.x & 31;
    const int wid   = threadIdx.x >> 5;
    const int strip = blockIdx.x * WAVES + wid;      // 0..3071
    const int plane = strip >> 5;                    // 0..95  (N*C planes)
    const int x0    = (strip & 31) << 4;             // 0..496

    const float* Pp = P + (size_t)plane * (IMG * IMG);
    const float* Tp = T + (size_t)plane * (IMG * IMG);
    float* sPw = &sP[wid][0];
    float* sTw = &sT[wid][0];
    float* sHw = &sH[wid][0];

    const int h = lane >> 4;   // lane half selects K sub-pair
    const int m = lane & 15;   // row (A) or column (B/D) index

    // ---- loop-invariant banded weight operands ------------------------------
    // Horizontal B (28x16): B[k][n] = w[k-n]; per K=4 chunk: k = 4c + 2h + {0,1}, n = m
    v2f Bh[7];
#pragma unroll
    for (int c = 0; c < 7; ++c) {
        int k = 4 * c + 2 * h;
        Bh[c][0] = band(k - m);
        Bh[c][1] = band(k + 1 - m);
    }
    // Vertical A (16x32): row m = output row; K indexes H rows (two 16-row groups,
    // duplicates at k=16..21 zero-weighted). k<16 -> tap k-m ; k>=22 -> tap k-m-6
    v2f Av[8];
#pragma unroll
    for (int c = 0; c < 8; ++c) {
        int k0 = 4 * c + 2 * h;
#pragma unroll
        for (int j = 0; j < 2; ++j) {
            int k = k0 + j;
            float w = (k >= 16 && k <= 21) ? 0.0f
                                           : band(k < 16 ? (k - m) : (k - m - 6));
            Av[c][j] = w;
        }
    }

    const float C1 = 0.0004f;  // (0.01*2)^2
    const float C2 = 0.0036f;  // (0.03*2)^2
    const v8f vzero = {0.f, 0.f, 0.f, 0.f, 0.f, 0.f, 0.f, 0.f};

    float lsum = 0.0f;

    for (int t = 0; t < 32; ++t) {
        const int y0 = t << 4;

        // ---- stage raw halo tiles (zero-padded) into LDS --------------------
        for (int r = 0; r < PROWS; ++r) {
            int gy = y0 - 5 + r;
            int gx = x0 - 5 + lane;
            float pv = 0.0f, tv = 0.0f;
            if (lane < PCOLS && gy >= 0 && gy < IMG && gx >= 0 && gx < IMG) {
                size_t o = (size_t)gy * IMG + gx;
                pv = Pp[o];
                tv = Tp[o];
            }
            if (lane < PCOLS) {
                sPw[r * PSTRIDE + lane] = pv;
                sTw[r * PSTRIDE + lane] = tv;
            }
        }
        lds_fence();

        // ---- horizontal pass: per group, 5 concurrent WMMA chains -----------
#pragma unroll
        for (int g = 0; g < 2; ++g) {
            const int row = m + 10 * g;   // raw LDS row for this A row
            float pa[14], ta[14];
#pragma unroll
            for (int c = 0; c < 7; ++c) {
                int base = row * PSTRIDE + 4 * c + 2 * h;
                pa[2 * c]     = sPw[base];
                pa[2 * c + 1] = sPw[base + 1];
                ta[2 * c]     = sTw[base];
                ta[2 * c + 1] = sTw[base + 1];
            }
            v8f hacc[5];
#pragma unroll
            for (int q = 0; q < 5; ++q) hacc[q] = vzero;

#pragma unroll
            for (int c = 0; c < 7; ++c) {
                float p0 = pa[2 * c], p1 = pa[2 * c + 1];
                float t0 = ta[2 * c], t1 = ta[2 * c + 1];
                v2f aq[5];
                aq[0][0] = p0;      aq[0][1] = p1;
                aq[1][0] = t0;      aq[1][1] = t1;
                aq[2][0] = p0 * p0; aq[2][1] = p1 * p1;
                aq[3][0] = t0 * t0; aq[3][1] = t1 * t1;
                aq[4][0] = p0 * t0; aq[4][1] = p1 * t1;
                // 5 independent accumulators -> no WMMA->WMMA RAW, pipe stays full
#pragma unroll
                for (int q = 0; q < 5; ++q)
                    hacc[q] = __builtin_amdgcn_wmma_f32_16x16x4_f32(
                        false, aq[q], false, Bh[c], (short)0, hacc[q], false, false);
            }
            // D layout -> H buffer rows 16g + v + 8h, col m (per quantity block)
#pragma unroll
            for (int q = 0; q < 5; ++q)
#pragma unroll
                for (int v = 0; v < 8; ++v)
                    sHw[q * HQ + (16 * g + v + 8 * h) * HSTRIDE + m] = hacc[q][v];
        }
        lds_fence();

        // ---- vertical pass: banded weights x H (K=32), 5 chains -------------
        v8f vac[5];
#pragma unroll
        for (int q = 0; q < 5; ++q) vac[q] = vzero;

#pragma unroll
        for (int c = 0; c < 8; ++c) {
            int k = 4 * c + 2 * h;
            v2f bq[5];
            // batch all 10 LDS reads for this chunk, then one wait, then 5 WMMAs
#pragma unroll
            for (int q = 0; q < 5; ++q) {
                bq[q][0] = sHw[q * HQ + k * HSTRIDE + m];
                bq[q][1] = sHw[q * HQ + (k + 1) * HSTRIDE + m];
            }
#pragma unroll
            for (int q = 0; q < 5; ++q)
                vac[q] = __builtin_amdgcn_wmma_f32_16x16x4_f32(
                    false, Av[c], false, bq[q], (short)0, vac[q], false, false);
        }
        lds_fence();   // WAR: next tile reuses sP/sT/sH

        // ---- fused SSIM map + partial mean ----------------------------------
#pragma unroll
        for (int v = 0; v < 8; ++v) {
            float mux = vac[0][v], muy = vac[1][v];
            float mux2 = mux * mux, muy2 = muy * muy, muxy = mux * muy;
            float sx  = fmaxf(vac[2][v] - mux2, 0.0f);
            float sy  = fmaxf(vac[3][v] - muy2, 0.0f);
            float sxy = vac[4][v] - muxy;
            float num = (2.0f * muxy + C1) * (2.0f * sxy + C2);
            float den = (mux2 + muy2 + C1) * (sx + sy + C2);
            lsum += num / den;
        }
    }

    // ---- wave32 reduce + one f64 atomic per wave ----------------------------
#pragma unroll
    for (int off = 16; off > 0; off >>= 1)
        lsum += __shfl_down(lsum, off, 32);
    if (lane == 0)
        atomicAdd(acc, (double)lsum);
}

__global__ void ssim_zero(double* a) {
    if (threadIdx.x == 0 && blockIdx.x == 0) a[0] = 0.0;
}

__global__ void ssim_finalize(const double* __restrict__ a,
                              float* __restrict__ out) {
    out[0] = 1.0f - (float)(a[0] / NPIX);
}

extern "C" void kernel_launch(void* const* d_in, const int* in_sizes, int n_in,
                              void* d_out, int out_size, void* d_ws, size_t ws_size,
                              hipStream_t stream) {
    const float* pred = (const float*)d_in[0];
    const float* targ = (const float*)d_in[1];
    double* acc = (double*)d_ws;          // 8 bytes of scratch for the f64 sum

    ssim_zero<<<1, 32, 0, stream>>>(acc);
    // 96 planes * 32 strips = 3072 strips; 1 wave per strip, 4 waves per block
    ssim_main<<<3072 / WAVES, WAVES * 32, 0, stream>>>(pred, targ, acc);
    ssim_finalize<<<1, 1, 0, stream>>>(acc, (float*)d_out);
}